// GANMainBlock_55705725829962
// MI455X (gfx1250) — compile-verified
//
#include <hip/hip_runtime.h>
#include <math.h>

#define HD 128
#define NEG_SLOPE 0.2f

typedef __attribute__((ext_vector_type(2))) float v2f;
typedef __attribute__((ext_vector_type(4))) float v4f;
typedef __attribute__((ext_vector_type(8))) float v8f;

__device__ __forceinline__ float leaky(float x) {
    return x > 0.0f ? x : NEG_SLOPE * x;
}

// ---------------------------------------------------------------------------
// Init workspace: agg = 0, seg_max = -inf, seg_sum = 0
// ---------------------------------------------------------------------------
__global__ __launch_bounds__(256) void gat_init_kernel(float* __restrict__ agg,
                                                       float* __restrict__ seg_max,
                                                       float* __restrict__ seg_sum,
                                                       int N) {
    size_t i = (size_t)blockIdx.x * blockDim.x + threadIdx.x;
    size_t total = (size_t)N * HD;
    if (i < total) agg[i] = 0.0f;
    if (i < (size_t)N) {
        seg_max[i] = -INFINITY;
        seg_sum[i] = 0.0f;
    }
}

// ---------------------------------------------------------------------------
// h = X @ W + b  via V_WMMA_F32_16X16X4_F32  (X: N x 128, W: 128 x 128)
// One block = 16 output rows. 8 waves/block, wave w owns column tile [16w,16w+16).
// Per wave: 32 chained WMMAs over K=128 in steps of 4.
//
// A (16x4, f32): lane 0-15  -> row M=lane,   {a.x,a.y} = K = {k, k+1}
//                lane 16-31 -> row M=lane-16,{a.x,a.y} = K = {k+2, k+3}
// B (4x16, f32): lane 0-15  -> col N=lane,   {b.x,b.y} = K rows {k, k+1}
//                lane 16-31 -> col N=lane-16,{b.x,b.y} = K rows {k+2, k+3}
// C/D (16x16):   VGPR r, lanes 0-15 = (M=r, N=lane); lanes 16-31 = (M=r+8, N=lane-16)
// ---------------------------------------------------------------------------
__global__ __launch_bounds__(256) void gat_gemm_h_kernel(const float* __restrict__ X,
                                                         const float* __restrict__ W,
                                                         const float* __restrict__ bias,
                                                         float* __restrict__ Hout,
                                                         int N) {
    const int wave = threadIdx.x >> 5;   // 0..7 -> column tile
    const int lane = threadIdx.x & 31;
    const int mlo  = lane & 15;
    const int hi   = lane >> 4;          // 0 or 1

    const int row0 = blockIdx.x * 16;
    const int n0   = wave * 16;

    // Clamp row for A loads (branchless; keeps EXEC all-ones for WMMA).
    int mrow = row0 + mlo;
    if (mrow > N - 1) mrow = N - 1;

    const float* Arow = X + (size_t)mrow * HD + 2 * hi;         // +k -> A[m][k + 2*hi]
    const float* Bcol = W + (size_t)(2 * hi) * HD + (n0 + mlo); // +k*HD -> W[k+2*hi][n]

    v8f acc = {};
#pragma unroll
    for (int k = 0; k < HD; k += 4) {
        v2f a = *(const v2f*)(Arow + k);          // 8B-aligned: global_load_b64
        v2f b;
        b.x = Bcol[(size_t)k * HD];
        b.y = Bcol[(size_t)k * HD + HD];
        acc = __builtin_amdgcn_wmma_f32_16x16x4_f32(
            /*neg_a=*/false, a, /*neg_b=*/false, b,
            /*c_mod=*/(short)0, acc, /*reuse_a=*/false, /*reuse_b=*/false);
    }

    const int ncol = n0 + mlo;
    const float bn = bias[ncol];
#pragma unroll
    for (int r = 0; r < 8; ++r) {
        int m = row0 + r + 8 * hi;
        if (m < N) Hout[(size_t)m * HD + ncol] = acc[r] + bn;
    }
}

// ---------------------------------------------------------------------------
// Per-edge logits + leaky_relu + segment max.  One wave (32 lanes) per edge,
// each lane handles 4 of 128 dims via 16B vectors (wave reads 512B contiguous).
// edge_attr is a 410 MB single-use stream -> non-temporal load so it doesn't
// evict the L2-resident h/agg/seg arrays (working set ~60 MB << 192 MB L2).
// ---------------------------------------------------------------------------
__global__ __launch_bounds__(256) void gat_edge_logits_kernel(
    const float* __restrict__ edge_attr,
    const int* __restrict__ edge_index,   // [2, E]
    const float* __restrict__ h,
    const float* __restrict__ W_edge,     // 384 x 1 : We | Ws | Wd
    const float* __restrict__ b_edge,
    float* __restrict__ e_out,
    float* __restrict__ seg_max,
    int E) {
    const int lane = threadIdx.x & 31;
    const int wid  = (int)(((size_t)blockIdx.x * blockDim.x + threadIdx.x) >> 5);
    if (wid >= E) return;

    const int src = edge_index[wid];
    const int dst = edge_index[(size_t)E + wid];

    const v4f we = ((const v4f*)W_edge)[lane];        // We[4l..4l+3]
    const v4f ws = ((const v4f*)W_edge)[32 + lane];   // Ws
    const v4f wd = ((const v4f*)W_edge)[64 + lane];   // Wd

    const v4f xe = __builtin_nontemporal_load(
        ((const v4f*)(edge_attr + (size_t)wid * HD)) + lane);   // TH=NT stream
    const v4f xs = ((const v4f*)(h + (size_t)src * HD))[lane];
    const v4f xd = ((const v4f*)(h + (size_t)dst * HD))[lane];

    float p = xe.x * we.x + xe.y * we.y + xe.z * we.z + xe.w * we.w
            + xs.x * ws.x + xs.y * ws.y + xs.z * ws.z + xs.w * ws.w
            + xd.x * wd.x + xd.y * wd.y + xd.z * wd.z + xd.w * wd.w;

#pragma unroll
    for (int off = 16; off > 0; off >>= 1) p += __shfl_xor(p, off, 32);

    if (lane == 0) {
        float e = leaky(p + b_edge[0]);
        e_out[wid] = e;
        atomicMax(&seg_max[src], e);   // global_atomic_max_num_f32
    }
}

// ---------------------------------------------------------------------------
// exp(e - seg_max[src]) in place + segment sum. Thread per edge.
// ---------------------------------------------------------------------------
__global__ __launch_bounds__(256) void gat_edge_softmax_kernel(
    const int* __restrict__ edge_index,
    const float* __restrict__ seg_max,
    float* __restrict__ seg_sum,
    float* __restrict__ e_buf,
    int E) {
    int i = (int)((size_t)blockIdx.x * blockDim.x + threadIdx.x);
    if (i >= E) return;
    int src = edge_index[i];
    float ex = __expf(e_buf[i] - seg_max[src]);
    e_buf[i] = ex;
    atomicAdd(&seg_sum[src], ex);
}

// ---------------------------------------------------------------------------
// agg[src] += (exp_e / seg_sum[src]) * h[dst].  One wave per edge; all targets
// are L2-resident (agg = 25.6 MB << 192 MB L2).
// ---------------------------------------------------------------------------
__global__ __launch_bounds__(256) void gat_edge_agg_kernel(
    const int* __restrict__ edge_index,
    const float* __restrict__ h,
    const float* __restrict__ e_buf,
    const float* __restrict__ seg_sum,
    float* __restrict__ agg,
    int E) {
    const int lane = threadIdx.x & 31;
    const int wid  = (int)(((size_t)blockIdx.x * blockDim.x + threadIdx.x) >> 5);
    if (wid >= E) return;

    const int src = edge_index[wid];
    const int dst = edge_index[(size_t)E + wid];
    const float a = e_buf[wid] / seg_sum[src];

    const v4f hd = ((const v4f*)(h + (size_t)dst * HD))[lane];
    float* arow = agg + (size_t)src * HD + lane * 4;
    atomicAdd(arow + 0, a * hd.x);
    atomicAdd(arow + 1, a * hd.y);
    atomicAdd(arow + 2, a * hd.z);
    atomicAdd(arow + 3, a * hd.w);
}

// ---------------------------------------------------------------------------
// out = leaky_relu(h + agg); output written once, never re-read -> NT store.
// ---------------------------------------------------------------------------
__global__ __launch_bounds__(256) void gat_finalize_kernel(const float* __restrict__ h,
                                                           const float* __restrict__ agg,
                                                           float* __restrict__ out,
                                                           size_t total) {
    size_t i = (size_t)blockIdx.x * blockDim.x + threadIdx.x;
    if (i < total) {
        __builtin_nontemporal_store(leaky(h[i] + agg[i]), out + i);
    }
}

// ---------------------------------------------------------------------------
extern "C" void kernel_launch(void* const* d_in, const int* in_sizes, int n_in,
                              void* d_out, int out_size, void* d_ws, size_t ws_size,
                              hipStream_t stream) {
    const float* node_attr  = (const float*)d_in[0];
    const float* edge_attr  = (const float*)d_in[1];
    const int*   edge_index = (const int*)d_in[2];
    // d_in[3] = num_nodes (device scalar, not host-readable under capture)
    const float* W_node = (const float*)d_in[4];
    const float* b_node = (const float*)d_in[5];
    const float* W_edge = (const float*)d_in[6];
    const float* b_edge = (const float*)d_in[7];

    const int N = in_sizes[0] / HD;
    const int E = in_sizes[2] / 2;

    float* out = (float*)d_out;

    // Workspace layout inside d_ws
    float* h       = (float*)d_ws;                 // N*128
    float* agg     = h + (size_t)N * HD;           // N*128
    float* seg_max = agg + (size_t)N * HD;         // N
    float* seg_sum = seg_max + N;                  // N
    float* e_buf   = seg_sum + N;                  // E

    const size_t totalNH = (size_t)N * HD;
    const int nhBlocks   = (int)((totalNH + 255) / 256);
    const int gemmBlocks = (N + 15) / 16;
    const int waveBlocks = (E + 7) / 8;            // 8 waves / 256-thread block
    const int eBlocks    = (E + 255) / 256;

    gat_init_kernel<<<nhBlocks, 256, 0, stream>>>(agg, seg_max, seg_sum, N);

    gat_gemm_h_kernel<<<gemmBlocks, 256, 0, stream>>>(node_attr, W_node, b_node, h, N);

    gat_edge_logits_kernel<<<waveBlocks, 256, 0, stream>>>(
        edge_attr, edge_index, h, W_edge, b_edge, e_buf, seg_max, E);

    gat_edge_softmax_kernel<<<eBlocks, 256, 0, stream>>>(
        edge_index, seg_max, seg_sum, e_buf, E);

    gat_edge_agg_kernel<<<waveBlocks, 256, 0, stream>>>(
        edge_index, h, e_buf, seg_sum, agg, E);

    gat_finalize_kernel<<<nhBlocks, 256, 0, stream>>>(h, agg, out, totalNH);
}